// MultiResCNNOutputLayer_75943611728730
// MI455X (gfx1250) — compile-verified
//
#include <hip/hip_runtime.h>
#include <math.h>

// Problem constants from the reference
#define B_  8
#define L_  2500
#define D_  512
#define Y_  8921

#define YT  64      // y-rows per block (4 waves x 16)
#define WY  16      // y-rows per wave (one WMMA M-tile)
#define LC  32      // l-columns per chunk (two WMMA N-tiles; 32 rows x 2KB = WGP$ size)

typedef __attribute__((ext_vector_type(2))) float v2f;
typedef __attribute__((ext_vector_type(8))) float v8f;

// Fused: S = U @ x^T, T = F @ x^T (shared A and B operands), online softmax
// over l, y[b,y] = (sum_l e^{S-m} * T) / (sum_l e^{S-m}) + bias[y]
__global__ __launch_bounds__(128) void attn_label_pool_kernel(
    const float* __restrict__ x,     // (B, L, D)
    const float* __restrict__ Uw,    // (Y, D)
    const float* __restrict__ Fw,    // (Y, D)
    const float* __restrict__ bias,  // (Y)
    float* __restrict__ out)         // (B, Y)
{
    // 256 KB of LDS: U/F tiles resident for the whole kernel (read from HBM once).
    __shared__ float U_lds[YT * D_];
    __shared__ float F_lds[YT * D_];

    const int b   = blockIdx.y;
    const int y0  = blockIdx.x * YT;
    const int tid = threadIdx.x;

    // ---- Stage 64x512 U and F tiles into LDS (rows clamped at Y-1 for the tail tile)
    for (int i = tid; i < YT * (D_ / 4); i += 128) {
        const int r  = i >> 7;          // / (D_/4)
        const int c4 = i & 127;         // % (D_/4)
        int gy = y0 + r; if (gy >= Y_) gy = Y_ - 1;
        const float4 u4 = *(const float4*)(Uw + (size_t)gy * D_ + c4 * 4);
        const float4 f4 = *(const float4*)(Fw + (size_t)gy * D_ + c4 * 4);
        *(float4*)(U_lds + r * D_ + c4 * 4) = u4;
        *(float4*)(F_lds + r * D_ + c4 * 4) = f4;
    }
    __syncthreads();

    const int wave = tid >> 5;
    const int lane = tid & 31;
    const int n    = lane & 15;   // A-matrix row m / B-matrix column n
    const int h    = lane >> 4;   // half-wave selector (K sub-pair / C row-half)

    // A-operand row bases in LDS (ISA 7.12.2: lane m holds A[m][2h], A[m][2h+1])
    const float* __restrict__ Ua = U_lds + (wave * WY + n) * D_;
    const float* __restrict__ Fa = F_lds + (wave * WY + n) * D_;

    // Online softmax state, replicated across the 16 lanes of each half-group.
    // Component j corresponds to y-row (wave*16 + 8*h + j).
    float run_m[8], run_z[8], run_a[8];
    #pragma unroll
    for (int j = 0; j < 8; ++j) {
        run_m[j] = -__builtin_inff();
        run_z[j] = 0.0f;
        run_a[j] = 0.0f;
    }

    const float* __restrict__ xb = x + (size_t)b * L_ * D_;

    for (int l0 = 0; l0 < L_; l0 += LC) {
        // Two B tiles: columns l0+n and l0+16+n (clamped row reads; masked later)
        int lr0 = l0 + n;       const bool cv0 = (lr0 < L_); if (lr0 >= L_) lr0 = L_ - 1;
        int lr1 = l0 + 16 + n;  const bool cv1 = (lr1 < L_); if (lr1 >= L_) lr1 = L_ - 1;
        const float* __restrict__ xr0 = xb + (size_t)lr0 * D_;
        const float* __restrict__ xr1 = xb + (size_t)lr1 * D_;

        // Prefetch next chunk's rows into the cache hierarchy (global_prefetch_b8)
        {
            int pl = l0 + LC + n; if (pl >= L_) pl = L_ - 1;
            __builtin_prefetch(xb + (size_t)pl * D_, 0, 3);
            int ph = l0 + LC + 16 + n; if (ph >= L_) ph = L_ - 1;
            __builtin_prefetch(xb + (size_t)ph * D_, 0, 3);
        }

        v8f s0 = {}, s1 = {};   // S tiles (16x16 f32 each)
        v8f t0 = {}, t1 = {};   // T tiles

        #pragma unroll 8
        for (int k = 0; k < D_; k += 4) {
            const int off = k + 2 * h;
            const v2f au  = *(const v2f*)(Ua + off);    // ds_load_b64
            const v2f af  = *(const v2f*)(Fa + off);    // ds_load_b64
            const v2f bx0 = *(const v2f*)(xr0 + off);   // global_load_b64 (WGP$-hit)
            const v2f bx1 = *(const v2f*)(xr1 + off);   // global_load_b64 (WGP$-hit)
            // Each A load feeds two WMMAs -> A-side DS bytes per FLOP halved
            s0 = __builtin_amdgcn_wmma_f32_16x16x4_f32(false, au, false, bx0, (short)0, s0, false, false);
            s1 = __builtin_amdgcn_wmma_f32_16x16x4_f32(false, au, false, bx1, (short)0, s1, false, false);
            t0 = __builtin_amdgcn_wmma_f32_16x16x4_f32(false, af, false, bx0, (short)0, t0, false, false);
            t1 = __builtin_amdgcn_wmma_f32_16x16x4_f32(false, af, false, bx1, (short)0, t1, false, false);
        }

        // Flash-style online softmax update across this 32-column chunk.
        // Combine the two tiles locally first, then one cross-lane reduction
        // (shfl-xor within the 16-lane half-group; masks 1,2,4,8).
        #pragma unroll
        for (int j = 0; j < 8; ++j) {
            const float sv0 = cv0 ? s0[j] : -__builtin_inff();
            const float sv1 = cv1 ? s1[j] : -__builtin_inff();

            float mx = fmaxf(sv0, sv1);
            mx = fmaxf(mx, __shfl_xor(mx, 1, 32));
            mx = fmaxf(mx, __shfl_xor(mx, 2, 32));
            mx = fmaxf(mx, __shfl_xor(mx, 4, 32));
            mx = fmaxf(mx, __shfl_xor(mx, 8, 32));

            const float nm = fmaxf(run_m[j], mx);      // finite after first chunk
            const float sc = __expf(run_m[j] - nm);    // 0 on first chunk (-inf arg)
            const float p0 = __expf(sv0 - nm);         // 0 for invalid columns
            const float p1 = __expf(sv1 - nm);

            float ps = p0 + p1;
            float pt = p0 * t0[j] + p1 * t1[j];
            ps += __shfl_xor(ps, 1, 32);  pt += __shfl_xor(pt, 1, 32);
            ps += __shfl_xor(ps, 2, 32);  pt += __shfl_xor(pt, 2, 32);
            ps += __shfl_xor(ps, 4, 32);  pt += __shfl_xor(pt, 4, 32);
            ps += __shfl_xor(ps, 8, 32);  pt += __shfl_xor(pt, 8, 32);

            run_z[j] = run_z[j] * sc + ps;
            run_a[j] = run_a[j] * sc + pt;
            run_m[j] = nm;
        }
    }

    // C/D layout: VGPR j holds row (8h + j); stats are uniform across each
    // half-group, so lane n==0 of each half writes its 8 rows.
    if (n == 0) {
        #pragma unroll
        for (int j = 0; j < 8; ++j) {
            const int yy = y0 + wave * WY + 8 * h + j;
            if (yy < Y_) {
                out[(size_t)b * Y_ + yy] = run_a[j] / run_z[j] + bias[yy];
            }
        }
    }
}

extern "C" void kernel_launch(void* const* d_in, const int* in_sizes, int n_in,
                              void* d_out, int out_size, void* d_ws, size_t ws_size,
                              hipStream_t stream) {
    const float* x    = (const float*)d_in[0];  // (B, L, D)
    const float* Uw   = (const float*)d_in[1];  // (Y, D)
    const float* Fw   = (const float*)d_in[2];  // (Y, D)
    const float* bias = (const float*)d_in[3];  // (Y)
    float* out = (float*)d_out;                 // (B, Y)

    dim3 grid((Y_ + YT - 1) / YT, B_);          // (140, 8)
    attn_label_pool_kernel<<<grid, 128, 0, stream>>>(x, Uw, Fw, bias, out);
}